// GNN4_38414187495972
// MI455X (gfx1250) — compile-verified
//
#include <hip/hip_runtime.h>
#include <math.h>

// ---------------------------------------------------------------------------
// GATv2 GNN forward for MI455X (gfx1250, wave32).
//   - fp32 WMMA (v_wmma_f32_16x16x4_f32) for all node-level GEMMs (exact math)
//   - lin_l/lin_r fused: one wave per 16-row strip computes all column tiles
//     for BOTH weight matrices -> A fragment loaded once per K-step, 12 WMMAs
//   - float4 (global_load_b128) gathers in the bandwidth-dominated edge passes
// ---------------------------------------------------------------------------

typedef __attribute__((ext_vector_type(2))) float v2f;
typedef __attribute__((ext_vector_type(8))) float v8f;

#define NEG_SLOPE 0.2f
#define BN_EPS 1e-5f

// ---------------- utility fills ----------------
__global__ void k_fill_f32(float* __restrict__ p, float v, int n) {
  int i = blockIdx.x * blockDim.x + threadIdx.x;
  if (i < n) p[i] = v;
}
__global__ void k_fill_u32(unsigned* __restrict__ p, unsigned v, int n) {
  int i = blockIdx.x * blockDim.x + threadIdx.x;
  if (i < n) p[i] = v;
}

// ---------------- encoders (K=3, scalar math, b128 stores) ----------------
__global__ void k_node_encode(const float* __restrict__ x, const float* __restrict__ W,
                              const float* __restrict__ b, float* __restrict__ h, int N) {
  int i = blockIdx.x * blockDim.x + threadIdx.x;   // one thread per (node, 4 cols)
  if (i >= N * 64) return;
  int n = i >> 6, j4 = (i & 63) * 4;
  float x0 = x[n * 3], x1 = x[n * 3 + 1], x2 = x[n * 3 + 2];
  float4 w0 = *(const float4*)(W + j4);
  float4 w1 = *(const float4*)(W + 256 + j4);
  float4 w2 = *(const float4*)(W + 512 + j4);
  float4 bb = *(const float4*)(b + j4);
  float4 r;
  r.x = x0 * w0.x + x1 * w1.x + x2 * w2.x + bb.x;
  r.y = x0 * w0.y + x1 * w1.y + x2 * w2.y + bb.y;
  r.z = x0 * w0.z + x1 * w1.z + x2 * w2.z + bb.z;
  r.w = x0 * w0.w + x1 * w1.w + x2 * w2.w + bb.w;
  *(float4*)(h + (size_t)n * 256 + j4) = r;
}

__global__ void k_edge_encode(const float* __restrict__ ea, const float* __restrict__ W,
                              const float* __restrict__ b, float* __restrict__ out, int E) {
  int i = blockIdx.x * blockDim.x + threadIdx.x;   // one thread per (edge, 4 cols)
  if (i >= E * 4) return;
  int e = i >> 2, j4 = (i & 3) * 4;
  float x0 = ea[e * 3], x1 = ea[e * 3 + 1], x2 = ea[e * 3 + 2];
  float4 w0 = *(const float4*)(W + j4);
  float4 w1 = *(const float4*)(W + 16 + j4);
  float4 w2 = *(const float4*)(W + 32 + j4);
  float4 bb = *(const float4*)(b + j4);
  float4 r;
  r.x = x0 * w0.x + x1 * w1.x + x2 * w2.x + bb.x;
  r.y = x0 * w0.y + x1 * w1.y + x2 * w2.y + bb.y;
  r.z = x0 * w0.z + x1 * w1.z + x2 * w2.z + bb.z;
  r.w = x0 * w0.w + x1 * w1.w + x2 * w2.w + bb.w;
  *(float4*)(out + (size_t)e * 16 + j4) = r;
}

// ---------------- self-loop edge attr: mean of incoming encoded attrs -------
__global__ void k_loop_acc(const float* __restrict__ eaf, const int* __restrict__ dst,
                           float* __restrict__ loop, int* __restrict__ cnt, int E) {
  int i = blockIdx.x * blockDim.x + threadIdx.x;
  if (i >= E * 16) return;
  int e = i >> 4, k = i & 15;
  int d = dst[e];
  atomicAdd(&loop[(size_t)d * 16 + k], eaf[i]);
  if (k == 0) atomicAdd(&cnt[d], 1);
}

__global__ void k_loop_div(float* __restrict__ loop, const int* __restrict__ cnt, int N) {
  int i = blockIdx.x * blockDim.x + threadIdx.x;
  if (i >= N * 16) return;
  int c = cnt[i >> 4];
  loop[i] /= (c > 1 ? (float)c : 1.0f);
}

// ---------------- fused fp32 WMMA GEMM -------------------------------------
// One wave per 16-row strip of A[M x K]; computes NT 16-col tiles of
//   Cl = A @ Wl + bl          (always)
//   Cr = A @ Wr + br          (if DUAL)   -- shares the A fragment
//   Cl = 0.5*Cl + 0.5*blend   (if BLEND)  -- skip connection
// Fragment layouts per CDNA5 ISA 7.12.2:
//   A 16x4 : lane(0-15)=M, VGPR0={K+0 | K+2 hi half}, VGPR1={K+1 | K+3}
//   B 4x16 : lane(0-15)=N, VGPR0=row{K+0 | K+2},      VGPR1=row{K+1 | K+3}
//   C 16x16: VGPR v, lanes 0-15 -> M=v, lanes 16-31 -> M=v+8; N=lane&15
template <int NT, bool DUAL, bool BLEND>
__global__ __launch_bounds__(32) void k_gat_gemm(
    const float* __restrict__ A, const float* __restrict__ Wl,
    const float* __restrict__ bl, float* __restrict__ Cl,
    const float* __restrict__ Wr, const float* __restrict__ br,
    float* __restrict__ Cr, const float* __restrict__ blend, int K) {
  constexpr int Nc = NT * 16;
  int row0 = blockIdx.x * 16;
  int lane = threadIdx.x;
  int half = lane >> 4;     // 0: lanes 0-15, 1: lanes 16-31
  int l = lane & 15;

  const float* arow = A + (size_t)(row0 + l) * K;   // 16B-aligned row (K % 4 == 0)

  v8f accL[NT];
  v8f accR[NT];
#pragma unroll
  for (int t = 0; t < NT; ++t) {
    accL[t] = (v8f){};
    if (DUAL) accR[t] = (v8f){};
  }

  for (int k = 0; k < K; k += 4) {
    int ka = k + half * 2;
    v2f a = *(const v2f*)(arow + ka);   // single 8-B load feeds up to 2*NT WMMAs
#pragma unroll
    for (int t = 0; t < NT; ++t) {
      const float* wc = Wl + t * 16 + l;
      v2f b;
      b.x = wc[(size_t)ka * Nc];
      b.y = wc[(size_t)(ka + 1) * Nc];
      accL[t] = __builtin_amdgcn_wmma_f32_16x16x4_f32(
          false, a, false, b, (short)0, accL[t], false, false);
      if (DUAL) {
        const float* wc2 = Wr + t * 16 + l;
        v2f b2;
        b2.x = wc2[(size_t)ka * Nc];
        b2.y = wc2[(size_t)(ka + 1) * Nc];
        accR[t] = __builtin_amdgcn_wmma_f32_16x16x4_f32(
            false, a, false, b2, (short)0, accR[t], false, false);
      }
    }
  }

#pragma unroll
  for (int t = 0; t < NT; ++t) {
    int col = t * 16 + l;
    float bvl = bl[col];
    float bvr = DUAL ? br[col] : 0.f;
#pragma unroll
    for (int v = 0; v < 8; ++v) {
      int r = row0 + v + half * 8;
      float val = accL[t][v] + bvl;
      if (BLEND) val = 0.5f * val + 0.5f * blend[(size_t)r * Nc + col];
      Cl[(size_t)r * Nc + col] = val;
      if (DUAL) Cr[(size_t)r * Nc + col] = accR[t][v] + bvr;
    }
  }
}

// ---------------- attention logits + running max ----------------
// one thread per (edge-with-self-loops, head). Channels == 16 always.
__global__ void k_logits(const float* __restrict__ xl, const float* __restrict__ xr,
                         const float* __restrict__ eaf, const int* __restrict__ src,
                         const int* __restrict__ dst, const float* __restrict__ lin_e,
                         const float* __restrict__ att, float* __restrict__ logit,
                         unsigned* __restrict__ amax, int E, int Nn, int H, int HC) {
  int i = blockIdx.x * blockDim.x + threadIdx.x;
  int total = (E + Nn) * H;
  if (i >= total) return;
  int e = i / H, h = i - e * H;
  int s, d;
  if (e < E) { s = src[e]; d = dst[e]; } else { s = e - E; d = s; }
  int hc0 = h * 16;

  // ev = ea_row @ lin_e[:, hc0:hc0+16]   (b128 loads throughout)
  float ev[16];
#pragma unroll
  for (int c = 0; c < 16; ++c) ev[c] = 0.f;
  const float4* ear4 = (const float4*)(eaf + (size_t)e * 16);
#pragma unroll
  for (int kk = 0; kk < 4; ++kk) {
    float4 ekv = ear4[kk];
    float eks[4] = { ekv.x, ekv.y, ekv.z, ekv.w };
#pragma unroll
    for (int kq = 0; kq < 4; ++kq) {
      int k = kk * 4 + kq;
      const float4* lrow4 = (const float4*)(lin_e + (size_t)k * HC + hc0);
#pragma unroll
      for (int cc = 0; cc < 4; ++cc) {
        float4 lw = lrow4[cc];
        ev[cc * 4 + 0] += eks[kq] * lw.x;
        ev[cc * 4 + 1] += eks[kq] * lw.y;
        ev[cc * 4 + 2] += eks[kq] * lw.z;
        ev[cc * 4 + 3] += eks[kq] * lw.w;
      }
    }
  }

  const float4* xls4 = (const float4*)(xl + (size_t)s * HC + hc0);
  const float4* xrd4 = (const float4*)(xr + (size_t)d * HC + hc0);
  const float4* att4 = (const float4*)(att + hc0);
  float acc = 0.f;
#pragma unroll
  for (int cc = 0; cc < 4; ++cc) {
    float4 av = xls4[cc];
    float4 bv = xrd4[cc];
    float4 wv = att4[cc];
    float m0 = av.x + bv.x + ev[cc * 4 + 0];
    float m1 = av.y + bv.y + ev[cc * 4 + 1];
    float m2 = av.z + bv.z + ev[cc * 4 + 2];
    float m3 = av.w + bv.w + ev[cc * 4 + 3];
    m0 = (m0 > 0.f) ? m0 : NEG_SLOPE * m0;
    m1 = (m1 > 0.f) ? m1 : NEG_SLOPE * m1;
    m2 = (m2 > 0.f) ? m2 : NEG_SLOPE * m2;
    m3 = (m3 > 0.f) ? m3 : NEG_SLOPE * m3;
    acc += m0 * wv.x + m1 * wv.y + m2 * wv.z + m3 * wv.w;
  }
  logit[(size_t)e * H + h] = acc;

  // order-preserving uint key for float atomic max (handles negatives)
  unsigned bits = __float_as_uint(acc);
  unsigned key = (bits & 0x80000000u) ? ~bits : (bits | 0x80000000u);
  atomicMax(&amax[(size_t)d * H + h], key);
}

// ---------------- exp-weighted aggregation (unnormalized) ----------------
__global__ void k_aggregate(const float* __restrict__ xl, const float* __restrict__ logit,
                            const unsigned* __restrict__ amax, const int* __restrict__ src,
                            const int* __restrict__ dst, float* __restrict__ asum,
                            float* __restrict__ agg, int E, int Nn, int H, int HC) {
  int i = blockIdx.x * blockDim.x + threadIdx.x;
  int total = (E + Nn) * H;
  if (i >= total) return;
  int e = i / H, h = i - e * H;
  int s, d;
  if (e < E) { s = src[e]; d = dst[e]; } else { s = e - E; d = s; }

  unsigned key = amax[(size_t)d * H + h];
  unsigned mb = (key & 0x80000000u) ? (key ^ 0x80000000u) : ~key;
  float mx = __uint_as_float(mb);
  float w = __expf(logit[(size_t)e * H + h] - mx);
  atomicAdd(&asum[(size_t)d * H + h], w);

  const float4* xls4 = (const float4*)(xl + (size_t)s * HC + h * 16);
  float* dp = agg + (size_t)d * HC + h * 16;
#pragma unroll
  for (int cc = 0; cc < 4; ++cc) {
    float4 xv = xls4[cc];
    atomicAdd(&dp[cc * 4 + 0], w * xv.x);
    atomicAdd(&dp[cc * 4 + 1], w * xv.y);
    atomicAdd(&dp[cc * 4 + 2], w * xv.z);
    atomicAdd(&dp[cc * 4 + 3], w * xv.w);
  }
}

// ---------------- normalize + bias + BatchNorm(eval) + ELU ----------------
__global__ void k_finalize(float* __restrict__ agg, const float* __restrict__ asum,
                           const float* __restrict__ bias, const float* __restrict__ gamma,
                           const float* __restrict__ beta, const float* __restrict__ rm,
                           const float* __restrict__ rv, int Nn, int D, int H) {
  int i = blockIdx.x * blockDim.x + threadIdx.x;
  if (i >= Nn * D) return;
  int n = i / D, j = i - n * D;
  int h = j >> 4;  // D == H*16 always
  float v = agg[i] / asum[(size_t)n * H + h] + bias[j];
  v = (v - rm[j]) * rsqrtf(rv[j] + BN_EPS) * gamma[j] + beta[j];
  agg[i] = (v > 0.f) ? v : expm1f(v);
}

// ---------------------------------------------------------------------------
extern "C" void kernel_launch(void* const* d_in, const int* in_sizes, int n_in,
                              void* d_out, int out_size, void* d_ws, size_t ws_size,
                              hipStream_t stream) {
  (void)n_in; (void)out_size; (void)ws_size;
  const int N = in_sizes[0] / 3;   // 50000
  const int E = in_sizes[1] / 2;   // 800000
  const int EF = E + N;

  const float* x      = (const float*)d_in[0];
  const int*   ei     = (const int*)d_in[1];
  const int*   srcI   = ei;
  const int*   dstI   = ei + E;
  const float* eattr  = (const float*)d_in[2];
  const float* node_w = (const float*)d_in[3];
  const float* node_b = (const float*)d_in[4];
  const float* edge_w = (const float*)d_in[5];
  const float* edge_b = (const float*)d_in[6];
  const float* skip_w = (const float*)d_in[7];
  const float* skip_b = (const float*)d_in[8];

  struct Layer {
    const float *lin_l, *b_l, *lin_r, *b_r, *lin_e, *att, *bias, *gamma, *beta, *rm, *rv;
  } L[3];
  for (int l = 0; l < 3; ++l) {
    int b = 9 + l * 11;
    L[l].lin_l = (const float*)d_in[b + 0];
    L[l].b_l   = (const float*)d_in[b + 1];
    L[l].lin_r = (const float*)d_in[b + 2];
    L[l].b_r   = (const float*)d_in[b + 3];
    L[l].lin_e = (const float*)d_in[b + 4];
    L[l].att   = (const float*)d_in[b + 5];
    L[l].bias  = (const float*)d_in[b + 6];
    L[l].gamma = (const float*)d_in[b + 7];
    L[l].beta  = (const float*)d_in[b + 8];
    L[l].rm    = (const float*)d_in[b + 9];
    L[l].rv    = (const float*)d_in[b + 10];
  }

  // workspace layout (floats; every section is a multiple of 4 floats -> 16B)
  float* ws = (float*)d_ws;
  size_t off = 0;
  float* h0    = ws + off; off += (size_t)N * 256;
  float* eaf   = ws + off; off += (size_t)EF * 16;   // [0,E): edges, [E,EF): self loops
  float* xl    = ws + off; off += (size_t)N * 96;
  float* xr    = ws + off; off += (size_t)N * 96;
  float* logit = ws + off; off += (size_t)EF * 6;
  unsigned* amax = (unsigned*)(ws + off); off += (size_t)N * 6;
  float* asum  = ws + off; off += (size_t)N * 6;
  float* out0  = ws + off; off += (size_t)N * 96;    // layer0 output (kept for skip)
  float* out1  = ws + off; off += (size_t)N * 96;
  float* out2  = ws + off; off += (size_t)N * 16;
  int*   cnt   = (int*)(ws + off); off += (size_t)N;

  const int TB = 256;
  auto nb = [](long n) { return (unsigned)((n + 255) / 256); };

  // encoders
  k_node_encode<<<nb((long)N * 64), TB, 0, stream>>>(x, node_w, node_b, h0, N);
  k_edge_encode<<<nb((long)E * 4), TB, 0, stream>>>(eattr, edge_w, edge_b, eaf, E);

  // self-loop edge attr = mean of incoming encoded attrs
  float* loopA = eaf + (size_t)E * 16;
  k_fill_f32<<<nb((long)N * 16), TB, 0, stream>>>(loopA, 0.f, N * 16);
  k_fill_u32<<<nb(N), TB, 0, stream>>>((unsigned*)cnt, 0u, N);
  k_loop_acc<<<nb((long)E * 16), TB, 0, stream>>>(eaf, dstI, loopA, cnt, E);
  k_loop_div<<<nb((long)N * 16), TB, 0, stream>>>(loopA, cnt, N);

  const float* hin[3] = { h0, out0, out1 };
  float* outb[3]      = { out0, out1, out2 };
  const int Kdim[3]   = { 256, 96, 96 };
  const int Hh[3]     = { 6, 6, 1 };

  const unsigned KEY_NEG_INF = 0x007FFFFFu;  // ordered key of -inf

  for (int l = 0; l < 3; ++l) {
    const int H = Hh[l];
    const int HC = H * 16;
    dim3 g((unsigned)(N / 16), 1);
    if (H == 6) {
      k_gat_gemm<6, true, false><<<g, 32, 0, stream>>>(
          hin[l], L[l].lin_l, L[l].b_l, xl, L[l].lin_r, L[l].b_r, xr, nullptr, Kdim[l]);
    } else {
      k_gat_gemm<1, true, false><<<g, 32, 0, stream>>>(
          hin[l], L[l].lin_l, L[l].b_l, xl, L[l].lin_r, L[l].b_r, xr, nullptr, Kdim[l]);
    }

    k_fill_u32<<<nb((long)N * H), TB, 0, stream>>>(amax, KEY_NEG_INF, N * H);
    k_logits<<<nb((long)EF * H), TB, 0, stream>>>(xl, xr, eaf, srcI, dstI,
                                                  L[l].lin_e, L[l].att, logit, amax,
                                                  E, N, H, HC);
    k_fill_f32<<<nb((long)N * H), TB, 0, stream>>>(asum, 0.f, N * H);
    k_fill_f32<<<nb((long)N * HC), TB, 0, stream>>>(outb[l], 0.f, N * HC);
    k_aggregate<<<nb((long)EF * H), TB, 0, stream>>>(xl, logit, amax, srcI, dstI,
                                                     asum, outb[l], E, N, H, HC);
    k_finalize<<<nb((long)N * HC), TB, 0, stream>>>(outb[l], asum, L[l].bias, L[l].gamma,
                                                    L[l].beta, L[l].rm, L[l].rv, N, HC, H);
  }

  // skip connection fused with final blend:
  // d_out = 0.5*out2 + 0.5*(out0 @ skip_w + skip_b)
  dim3 gs((unsigned)(N / 16), 1);
  k_gat_gemm<1, false, true><<<gs, 32, 0, stream>>>(
      out0, skip_w, skip_b, (float*)d_out, nullptr, nullptr, nullptr, out2, 96);
}